// TopologyAdaptiveMask_41351945125994
// MI455X (gfx1250) — compile-verified
//
#include <hip/hip_runtime.h>
#include <hip/hip_bf16.h>
#include <math.h>

// Problem constants (fixed by the reference).
#define N_NODES 10000
#define D_FEAT  256
#define E_EDGES 320000
#define THRESH  0.5f
#define EPSN    1e-8f

// Radix sort config: 4-bit digits, stable LSD.
#define TILE    4096                 // 256 threads x 16 elements
#define NB      ((E_EDGES + TILE - 1) / TILE)   // 79 tiles
#define RADIX   16

typedef float v2f __attribute__((ext_vector_type(2)));
typedef float v8f __attribute__((ext_vector_type(8)));

// ---------------------------------------------------------------- init
__global__ void init_out(float* __restrict__ out, int n) {
  int i = blockIdx.x * 256 + threadIdx.x;
  if (i < n) out[i] = -1.0f;
}

__global__ void init_cnt(unsigned* __restrict__ c) {
  if (threadIdx.x < 8) c[threadIdx.x] = 0u;
}

// ------------------------------------------------------ node inv-norms
// One wave per node; coalesced loads, wave32 shuffle reduction.
__global__ void node_norms(const float* __restrict__ x, float* __restrict__ invn) {
  int wave = threadIdx.x >> 5, lane = threadIdx.x & 31;
  int node = blockIdx.x * 8 + wave;
  if (node >= N_NODES) return;
  const float* p = x + (size_t)node * D_FEAT;
  float s = 0.f;
#pragma unroll
  for (int j = 0; j < D_FEAT / 32; ++j) { float v = p[lane + 32 * j]; s += v * v; }
#pragma unroll
  for (int off = 16; off; off >>= 1) s += __shfl_xor(s, off, 32);
  if (lane == 0) invn[node] = 1.0f / fmaxf(sqrtf(s), EPSN);
}

// --------------------------------------------- edge cosine sim via WMMA
// One wave computes 16 edges. A = xr[16x4], B = xc[4x16]; diagonal of the
// 16x16 f32 accumulator holds dot(x[row_i], x[col_i]) for edge i.
// Per ISA 7.12.2 (32-bit A 16x4 and B 4x16): lanes 0-15 hold K={k0,k0+1},
// lanes 16-31 hold K={k0+2,k0+3} -> per-lane contiguous float2 load.
__global__ void edge_sim_wmma(const float* __restrict__ x,
                              const int* __restrict__ eidx,
                              const float* __restrict__ invn,
                              float* __restrict__ sim) {
  int wave = threadIdx.x >> 5, lane = threadIdx.x & 31;
  int base = (blockIdx.x * 8 + wave) * 16;
  int sub  = lane & 15;
  int kq   = (lane >> 4) * 2;
  int r = eidx[base + sub];
  int c = eidx[E_EDGES + base + sub];
  const float* xr = x + (size_t)r * D_FEAT + kq;
  const float* xc = x + (size_t)c * D_FEAT + kq;
  v8f acc = {0.f, 0.f, 0.f, 0.f, 0.f, 0.f, 0.f, 0.f};
#pragma unroll 8
  for (int k0 = 0; k0 < D_FEAT; k0 += 4) {
    float2 af = *reinterpret_cast<const float2*>(xr + k0);
    float2 bf = *reinterpret_cast<const float2*>(xc + k0);
    v2f a; a[0] = af.x; a[1] = af.y;
    v2f b; b[0] = bf.x; b[1] = bf.y;
    acc = __builtin_amdgcn_wmma_f32_16x16x4_f32(false, a, false, b,
                                                (short)0, acc, false, false);
  }
  // D[sub][sub] at lanes 0..7 -> edges 0..7 (vgpr=sub); off-diagonal D[8+sub][sub] at lanes 16..23 ->
  // edges 8..15 (vgpr=sub).  Divergence only after the WMMAs (EXEC all-1s).
  if (sub < 8) {
    int di = (lane >> 4) * 8 + sub;
    float d = acc[0];
    d = sub == 1 ? acc[1] : d; d = sub == 2 ? acc[2] : d;
    d = sub == 3 ? acc[3] : d; d = sub == 4 ? acc[4] : d;
    d = sub == 5 ? acc[5] : d; d = sub == 6 ? acc[6] : d;
    d = sub == 7 ? acc[7] : d;
    int e  = base + di;
    int rr = eidx[e], cc = eidx[E_EDGES + e];
    sim[e] = d * invn[rr] * invn[cc];
  }
}

// -------------------------------------------------- keep mask + sort keys
__device__ __forceinline__ unsigned fkey_desc(float f) {
  unsigned u = __float_as_uint(f);
  u = (u >> 31) ? ~u : (u | 0x80000000u);  // monotone ascending f32 -> u32
  return ~u;                               // descending order key
}

__global__ void build_keep_keys(const float* __restrict__ sim,
                                unsigned* __restrict__ keep,
                                unsigned* __restrict__ recov,
                                unsigned* __restrict__ key,
                                unsigned* __restrict__ val,
                                unsigned* __restrict__ cnt, int n) {
  int i = blockIdx.x * 256 + threadIdx.x;
  if (i >= n) return;
  float s = sim[i];
  unsigned k = (s >= THRESH) ? 1u : 0u;
  keep[i]  = k;
  recov[i] = 0u;
  key[i]   = k ? 0xFFFFFFFFu : fkey_desc(s);  // kept edges sort last
  val[i]   = (unsigned)i;
  if (k) atomicAdd(&cnt[0], 1u);              // integer sum: deterministic
}

__global__ void compute_nrec(unsigned* __restrict__ cnt, int n) {
  if (threadIdx.x == 0 && blockIdx.x == 0) {
    unsigned npr = (unsigned)n - cnt[0];
    const float qtau = (float)(0.7 * (1.0 - 1.0 / 500.0));  // Q * tau = 0.6986
    cnt[1] = (unsigned)floorf(qtau * (float)npr);
  }
}

// ------------------------------------------- stable LSD radix sort (4-bit)
__global__ void radix_hist(const unsigned* __restrict__ key,
                           unsigned* __restrict__ gh, int shift, int n) {
  __shared__ unsigned cnt[RADIX];
  int t = threadIdx.x;
  if (t < RADIX) cnt[t] = 0u;
  __syncthreads();
  int base = blockIdx.x * TILE + t * 16;
  for (int j = 0; j < 16; ++j) {
    int i = base + j;
    if (i < n) atomicAdd(&cnt[(key[i] >> shift) & (RADIX - 1)], 1u);
  }
  __syncthreads();
  if (t < RADIX) gh[t * gridDim.x + blockIdx.x] = cnt[t];  // digit-major
}

__global__ void scan_excl(unsigned* __restrict__ a, int len) {
  __shared__ unsigned part[256], ps[256];
  int t = threadIdx.x;
  int chunk = (len + 255) >> 8;
  int s = t * chunk;
  int e = (s + chunk < len) ? s + chunk : len;
  unsigned sum = 0;
  for (int i = s; i < e; ++i) sum += a[i];
  part[t] = sum;
  __syncthreads();
  if (t == 0) {
    unsigned run = 0;
    for (int i = 0; i < 256; ++i) { unsigned v = part[i]; ps[i] = run; run += v; }
  }
  __syncthreads();
  unsigned run = ps[t];
  for (int i = s; i < e; ++i) { unsigned v = a[i]; a[i] = run; run += v; }
}

__global__ void radix_scatter(const unsigned* __restrict__ kin,
                              const unsigned* __restrict__ vin,
                              unsigned* __restrict__ kout,
                              unsigned* __restrict__ vout,
                              const unsigned* __restrict__ gh, int shift, int n) {
  __shared__ unsigned short tc[256 * RADIX];  // per-thread digit counts (8KB)
  __shared__ unsigned dbase[RADIX];
  int t = threadIdx.x;
#pragma unroll
  for (int d = 0; d < RADIX; ++d) tc[t * RADIX + d] = 0;
  if (t < RADIX) dbase[t] = gh[t * gridDim.x + blockIdx.x];
  int base = blockIdx.x * TILE + t * 16;
  for (int j = 0; j < 16; ++j) {                 // stable: thread-sequential
    int i = base + j;
    if (i < n) tc[t * RADIX + ((kin[i] >> shift) & (RADIX - 1))]++;
  }
  __syncthreads();
  if (t < RADIX) {                               // cross-thread exclusive scan
    unsigned run = 0;
    for (int tt = 0; tt < 256; ++tt) {
      unsigned v = tc[tt * RADIX + t];
      tc[tt * RADIX + t] = (unsigned short)run;
      run += v;
    }
  }
  __syncthreads();
  for (int j = 0; j < 16; ++j) {
    int i = base + j;
    if (i < n) {
      unsigned k = kin[i];
      unsigned d = (k >> shift) & (RADIX - 1);
      unsigned off = tc[t * RADIX + d];
      tc[t * RADIX + d] = (unsigned short)(off + 1);
      unsigned pos = dbase[d] + off;
      kout[pos] = k;
      vout[pos] = vin[i];
    }
  }
}

// -------------------------------------------------- recovery + imp keys
__global__ void mark_recovered(const unsigned* __restrict__ val,
                               const unsigned* __restrict__ cnt,
                               unsigned* __restrict__ recov, int n) {
  int i = blockIdx.x * 256 + threadIdx.x;
  if (i < n && (unsigned)i < cnt[1]) recov[val[i]] = 1u;
}

__global__ void build_imp_keys(const int* __restrict__ eidx,
                               const unsigned* __restrict__ keep,
                               const unsigned* __restrict__ recov,
                               unsigned* __restrict__ key,
                               unsigned* __restrict__ val, int n) {
  int i = blockIdx.x * 256 + threadIdx.x;
  if (i >= n) return;
  unsigned imp = keep[i] | recov[i];
  unsigned lin = (unsigned)eidx[i] * (unsigned)N_NODES + (unsigned)eidx[E_EDGES + i];
  key[i] = imp ? lin : (unsigned)N_NODES * (unsigned)N_NODES;  // 1e8 < 2^27
  val[i] = (unsigned)i;
}

__global__ void write_imp(const int* __restrict__ eidx,
                          const unsigned* __restrict__ key,
                          const unsigned* __restrict__ val,
                          float* __restrict__ out, int n) {
  int j = blockIdx.x * 256 + threadIdx.x;
  if (j >= n) return;
  if (key[j] < (unsigned)N_NODES * (unsigned)N_NODES) {
    unsigned e = val[j];
    out[2 * E_EDGES + j] = (float)eidx[e];
    out[3 * E_EDGES + j] = (float)eidx[E_EDGES + e];
  }
}

// -------------------------------------------------------- compaction
__global__ void build_mask(const unsigned* __restrict__ keep,
                           const unsigned* __restrict__ recov,
                           unsigned* __restrict__ mask, int mode, int n) {
  int i = blockIdx.x * 256 + threadIdx.x;
  if (i >= n) return;
  mask[i] = (mode == 0) ? keep[i] : ((keep[i] | recov[i]) ^ 1u);
}

__global__ void tile_sums(const unsigned* __restrict__ mask,
                          unsigned* __restrict__ tsum, int n) {
  __shared__ unsigned sh[256];
  int t = threadIdx.x;
  int base = blockIdx.x * TILE + t * 16;
  unsigned s = 0;
  for (int j = 0; j < 16; ++j) { int i = base + j; if (i < n) s += mask[i]; }
  sh[t] = s;
  __syncthreads();
  for (int off = 128; off; off >>= 1) {
    if (t < off) sh[t] += sh[t + off];
    __syncthreads();
  }
  if (t == 0) tsum[blockIdx.x] = sh[0];
}

__global__ void compact_scatter(const int* __restrict__ eidx,
                                const unsigned* __restrict__ mask,
                                const unsigned* __restrict__ tbase,
                                float* __restrict__ out0,
                                float* __restrict__ out1, int n) {
  __shared__ unsigned short ts[256];
  __shared__ unsigned ex[256];
  int t = threadIdx.x;
  int base = blockIdx.x * TILE + t * 16;
  unsigned s = 0;
  for (int j = 0; j < 16; ++j) { int i = base + j; if (i < n) s += mask[i]; }
  ts[t] = (unsigned short)s;
  __syncthreads();
  if (t == 0) {
    unsigned run = 0;
    for (int tt = 0; tt < 256; ++tt) { unsigned v = ts[tt]; ex[tt] = run; run += v; }
  }
  __syncthreads();
  unsigned pos = tbase[blockIdx.x] + ex[t];
  for (int j = 0; j < 16; ++j) {
    int i = base + j;
    if (i < n && mask[i]) {
      out0[pos] = (float)eidx[i];
      out1[pos] = (float)eidx[E_EDGES + i];
      ++pos;
    }
  }
}

// ================================================================ launch
extern "C" void kernel_launch(void* const* d_in, const int* in_sizes, int n_in,
                              void* d_out, int out_size, void* d_ws, size_t ws_size,
                              hipStream_t stream) {
  const float* x    = (const float*)d_in[0];
  const int*   eidx = (const int*)d_in[1];
  float*       out  = (float*)d_out;

  char* ws = (char*)d_ws;
  size_t off = 0;
  auto alloc = [&](size_t bytes) -> void* {
    void* p = ws + off;
    off += (bytes + 255) & ~(size_t)255;
    return p;
  };
  float*    invn  = (float*)alloc((size_t)N_NODES * 4);
  float*    sim   = (float*)alloc((size_t)E_EDGES * 4);
  unsigned* keep  = (unsigned*)alloc((size_t)E_EDGES * 4);
  unsigned* recov = (unsigned*)alloc((size_t)E_EDGES * 4);
  unsigned* keyA  = (unsigned*)alloc((size_t)E_EDGES * 4);
  unsigned* valA  = (unsigned*)alloc((size_t)E_EDGES * 4);
  unsigned* keyB  = (unsigned*)alloc((size_t)E_EDGES * 4);
  unsigned* valB  = (unsigned*)alloc((size_t)E_EDGES * 4);
  unsigned* mask  = (unsigned*)alloc((size_t)E_EDGES * 4);
  unsigned* gh    = (unsigned*)alloc((size_t)RADIX * NB * 4);
  unsigned* tsum  = (unsigned*)alloc((size_t)NB * 4);
  unsigned* cnt   = (unsigned*)alloc(64);

  const int EB = (E_EDGES + 255) / 256;  // 1250

  init_out<<<(6 * E_EDGES + 255) / 256, 256, 0, stream>>>(out, 6 * E_EDGES);
  init_cnt<<<1, 32, 0, stream>>>(cnt);
  node_norms<<<(N_NODES + 7) / 8, 256, 0, stream>>>(x, invn);
  edge_sim_wmma<<<E_EDGES / 128, 256, 0, stream>>>(x, eidx, invn, sim);
  build_keep_keys<<<EB, 256, 0, stream>>>(sim, keep, recov, keyA, valA, cnt, E_EDGES);
  compute_nrec<<<1, 1, 0, stream>>>(cnt, E_EDGES);

  // Sort 1: pruned sims descending (stable), 8 passes, ends in keyA/valA.
  for (int p = 0; p < 8; ++p) {
    unsigned* ki = (p & 1) ? keyB : keyA;
    unsigned* vi = (p & 1) ? valB : valA;
    unsigned* ko = (p & 1) ? keyA : keyB;
    unsigned* vo = (p & 1) ? valA : valB;
    int shift = p * 4;
    radix_hist<<<NB, 256, 0, stream>>>(ki, gh, shift, E_EDGES);
    scan_excl<<<1, 256, 0, stream>>>(gh, RADIX * NB);
    radix_scatter<<<NB, 256, 0, stream>>>(ki, vi, ko, vo, gh, shift, E_EDGES);
  }
  mark_recovered<<<EB, 256, 0, stream>>>(valA, cnt, recov, E_EDGES);

  // Sort 2: coalesce A_imp by row*N+col (keys < 2^27 -> 7 passes), ends in B.
  build_imp_keys<<<EB, 256, 0, stream>>>(eidx, keep, recov, keyA, valA, E_EDGES);
  for (int p = 0; p < 7; ++p) {
    unsigned* ki = (p & 1) ? keyB : keyA;
    unsigned* vi = (p & 1) ? valB : valA;
    unsigned* ko = (p & 1) ? keyA : keyB;
    unsigned* vo = (p & 1) ? valA : valB;
    int shift = p * 4;
    radix_hist<<<NB, 256, 0, stream>>>(ki, gh, shift, E_EDGES);
    scan_excl<<<1, 256, 0, stream>>>(gh, RADIX * NB);
    radix_scatter<<<NB, 256, 0, stream>>>(ki, vi, ko, vo, gh, shift, E_EDGES);
  }
  write_imp<<<EB, 256, 0, stream>>>(eidx, keyB, valB, out, E_EDGES);

  // A_pur: stable compaction of kept edges.
  build_mask<<<EB, 256, 0, stream>>>(keep, recov, mask, 0, E_EDGES);
  tile_sums<<<NB, 256, 0, stream>>>(mask, tsum, E_EDGES);
  scan_excl<<<1, 256, 0, stream>>>(tsum, NB);
  compact_scatter<<<NB, 256, 0, stream>>>(eidx, mask, tsum, out, out + E_EDGES, E_EDGES);

  // discarded: stable compaction of !keep & !recovered.
  build_mask<<<EB, 256, 0, stream>>>(keep, recov, mask, 1, E_EDGES);
  tile_sums<<<NB, 256, 0, stream>>>(mask, tsum, E_EDGES);
  scan_excl<<<1, 256, 0, stream>>>(tsum, NB);
  compact_scatter<<<NB, 256, 0, stream>>>(eidx, mask, tsum,
                                          out + 4 * E_EDGES, out + 5 * E_EDGES, E_EDGES);
}